// MyLossFunction_33852932227152
// MI455X (gfx1250) — compile-verified
//
#include <hip/hip_runtime.h>
#include <hip/hip_bf16.h>

typedef __attribute__((ext_vector_type(2))) float v2f;
typedef __attribute__((ext_vector_type(8))) float v8f;
typedef __attribute__((ext_vector_type(4))) int v4i;

typedef __attribute__((address_space(1))) v4i gv4i;
typedef __attribute__((address_space(3))) v4i lv4i;

#define ITEMS_PER_BLOCK 4
#define NTHREADS 128
#define NT 8
#define NR 2
#define KDIM 100

#if defined(__has_builtin)
#if __has_builtin(__builtin_amdgcn_global_load_async_to_lds_b128)
#define HAS_ASYNC_LDS 1
#endif
#if __has_builtin(__builtin_amdgcn_s_wait_asynccnt)
#define HAS_WAIT_ASYNC 1
#endif
#endif

struct Cx { float r, i; };
__device__ __forceinline__ Cx cmul(Cx a, Cx b)       { return {a.r*b.r - a.i*b.i, a.r*b.i + a.i*b.r}; }
__device__ __forceinline__ Cx cmul_bconj(Cx a, Cx b) { return {a.r*b.r + a.i*b.i, a.i*b.r - a.r*b.i}; } // a*conj(b)
__device__ __forceinline__ Cx cmul_aconj(Cx a, Cx b) { return {a.r*b.r + a.i*b.i, a.r*b.i - a.i*b.r}; } // conj(a)*b
__device__ __forceinline__ Cx cadd(Cx a, Cx b)       { return {a.r + b.r, a.i + b.i}; }
__device__ __forceinline__ Cx csub(Cx a, Cx b)       { return {a.r - b.r, a.i - b.i}; }

// A = phi(2x8) @ F(8x2), complex. F stored as Fr/Fi flat [t*2+s].
__device__ __forceinline__ void phi_times_F(const Cx ph[2][8], const float* Fr, const float* Fi,
                                            Cx M[2][2]) {
#pragma unroll
  for (int n = 0; n < 2; ++n)
#pragma unroll
    for (int s = 0; s < 2; ++s) {
      Cx acc = {0.f, 0.f};
#pragma unroll
      for (int t = 0; t < 8; ++t) {
        Cx f = {Fr[t*2 + s], Fi[t*2 + s]};
        acc = cadd(acc, cmul(ph[n][t], f));
      }
      M[n][s] = acc;
    }
}

// rate = Re(log(det(1 + A^H Mu^{-1} A))), Mu = C C^H + noise (elementwise)
__device__ __forceinline__ float rate2(const Cx A[2][2], const Cx C[2][2]) {
  const float NOISEc = 1e-12f;
  Cx Mu[2][2];
#pragma unroll
  for (int n = 0; n < 2; ++n)
#pragma unroll
    for (int m = 0; m < 2; ++m) {
      Cx t = {NOISEc, 0.f};  // noise is added to ALL entries elementwise
#pragma unroll
      for (int s = 0; s < 2; ++s) t = cadd(t, cmul_bconj(C[n][s], C[m][s]));
      Mu[n][m] = t;
    }
  Cx det = csub(cmul(Mu[0][0], Mu[1][1]), cmul(Mu[0][1], Mu[1][0]));
  float inv2 = 1.f / (det.r*det.r + det.i*det.i);
  Cx idet = {det.r * inv2, -det.i * inv2};
  Cx n01 = {-Mu[0][1].r, -Mu[0][1].i};
  Cx n10 = {-Mu[1][0].r, -Mu[1][0].i};
  Cx Mi[2][2];
  Mi[0][0] = cmul(Mu[1][1], idet);
  Mi[0][1] = cmul(n01, idet);
  Mi[1][0] = cmul(n10, idet);
  Mi[1][1] = cmul(Mu[0][0], idet);
  Cx T[2][2];
#pragma unroll
  for (int n = 0; n < 2; ++n)
#pragma unroll
    for (int s = 0; s < 2; ++s)
      T[n][s] = cadd(cmul(Mi[n][0], A[0][s]), cmul(Mi[n][1], A[1][s]));
  Cx P[2][2];
#pragma unroll
  for (int r = 0; r < 2; ++r)
#pragma unroll
    for (int s = 0; s < 2; ++s) {
      P[r][s] = cadd(cmul_aconj(A[0][r], T[0][s]), cmul_aconj(A[1][r], T[1][s]));
      P[r][s].r += 1.f;  // 1 + inner, elementwise (faithful to reference)
    }
  Cx dp = csub(cmul(P[0][0], P[1][1]), cmul(P[0][1], P[1][0]));
  // Re(log z) = 0.5*log(|z|^2)
  return 0.5f * logf(dp.r*dp.r + dp.i*dp.i);
}

// global -> LDS staging: async (no VGPR round-trip, tracked by ASYNCcnt) if available
__device__ __forceinline__ void stage_f4(float* dst, const float* __restrict__ src,
                                         int nvec4, int tid) {
#ifdef HAS_ASYNC_LDS
  for (int i = tid; i < nvec4; i += NTHREADS) {
    __builtin_amdgcn_global_load_async_to_lds_b128(
        (gv4i*)(src + 4*i), (lv4i*)(dst + 4*i), 0, 0);
  }
#else
  const float4* s = reinterpret_cast<const float4*>(src);
  float4* d = reinterpret_cast<float4*>(dst);
  for (int i = tid; i < nvec4; i += NTHREADS) d[i] = s[i];
#endif
}

__device__ __forceinline__ void stage_fence() {
#ifdef HAS_ASYNC_LDS
#ifdef HAS_WAIT_ASYNC
  __builtin_amdgcn_s_wait_asynccnt(0);
#else
  asm volatile("s_wait_asynccnt 0x0" ::: "memory");
#endif
#endif
}

__global__ void __launch_bounds__(NTHREADS)
rate_loss_kernel(const float* __restrict__ theta1,
                 const float* __restrict__ g1r, const float* __restrict__ g1i,
                 const float* __restrict__ g2r, const float* __restrict__ g2i,
                 const float* __restrict__ ur,  const float* __restrict__ ui,
                 float* __restrict__ block_sums) {
  __shared__ float s_theta[ITEMS_PER_BLOCK][264];
  __shared__ float s_g1r[ITEMS_PER_BLOCK][2*KDIM];
  __shared__ float s_g1i[ITEMS_PER_BLOCK][2*KDIM];
  __shared__ float s_g2r[ITEMS_PER_BLOCK][2*KDIM];
  __shared__ float s_g2i[ITEMS_PER_BLOCK][2*KDIM];
  __shared__ float s_ur[ITEMS_PER_BLOCK][KDIM*NT];
  __shared__ float s_ui[ITEMS_PER_BLOCK][KDIM*NT];
  __shared__ float s_D[ITEMS_PER_BLOCK][8*16];
  __shared__ float s_red[ITEMS_PER_BLOCK];

  const int tid = threadIdx.x;
  const size_t b0 = (size_t)blockIdx.x * ITEMS_PER_BLOCK;

  // ---- Stage the 4 items' contiguous rows into LDS (async b128 global->LDS) ----
  stage_f4(&s_theta[0][0], theta1 + b0*264, (ITEMS_PER_BLOCK*264)/4, tid);
  stage_f4(&s_g1r[0][0],   g1r + b0*2*KDIM, (ITEMS_PER_BLOCK*2*KDIM)/4, tid);
  stage_f4(&s_g1i[0][0],   g1i + b0*2*KDIM, (ITEMS_PER_BLOCK*2*KDIM)/4, tid);
  stage_f4(&s_g2r[0][0],   g2r + b0*2*KDIM, (ITEMS_PER_BLOCK*2*KDIM)/4, tid);
  stage_f4(&s_g2i[0][0],   g2i + b0*2*KDIM, (ITEMS_PER_BLOCK*2*KDIM)/4, tid);
  stage_f4(&s_ur[0][0],    ur + b0*KDIM*NT, (ITEMS_PER_BLOCK*KDIM*NT)/4, tid);
  stage_f4(&s_ui[0][0],    ui + b0*KDIM*NT, (ITEMS_PER_BLOCK*KDIM*NT)/4, tid);
  stage_fence();     // drain this wave's ASYNCcnt
  __syncthreads();   // make all waves' staged data visible

  const int w    = tid >> 5;   // wave -> item
  const int lane = tid & 31;
  const int m    = lane & 15;  // A row / B col index
  const int grp  = lane >> 4;  // k sub-group

  const float* th = &s_theta[w][0];

  // A-matrix row selection: rows 0-1 Re(G1d), 2-3 Im(G1d), 4-5 Re(G2d), 6-7 Im(G2d)
  const bool act   = (m < 8);
  const int  mm    = act ? m : 0;
  const int  useG2 = (mm >> 2) & 1;
  const int  useIm = (mm >> 1) & 1;
  const int  row   = mm & 1;
  const float amask = act ? 1.f : 0.f;
  const float* gr = (useG2 ? &s_g2r[w][0] : &s_g1r[w][0]) + row * KDIM;
  const float* gi = (useG2 ? &s_g2i[w][0] : &s_g1i[w][0]) + row * KDIM;
  // B-matrix column: cols 0-7 Re(U), cols 8-15 Im(U); U is [k][t], stride 8
  const float* uc = (m < 8) ? (&s_ur[w][0] + m) : (&s_ui[w][0] + (m - 8));

  // ---- D(16x16) = A(16x100) * B(100x16) via 25x V_WMMA_F32_16X16X4_F32 ----
  v8f acc0 = {0.f,0.f,0.f,0.f,0.f,0.f,0.f,0.f};
  v8f acc1 = {0.f,0.f,0.f,0.f,0.f,0.f,0.f,0.f};
  for (int k0 = 0; k0 < KDIM; k0 += 8) {
#pragma unroll
    for (int half = 0; half < 2; ++half) {
      const int kc = k0 + 4*half;
      if (kc >= KDIM) break;
      const int kb = kc + 2*grp;
      v2f a, b;
#pragma unroll
      for (int r = 0; r < 2; ++r) {
        const int k = kb + r;
        const float tr = th[k], ti = th[KDIM + k];
        const float gvr = gr[k], gvi = gi[k];
        const float re  = gvr*tr - gvi*ti;
        const float im  = gvr*ti + gvi*tr;
        a[r] = amask * (useIm ? im : re);
        b[r] = uc[k*NT];
      }
      if (half == 0)
        acc0 = __builtin_amdgcn_wmma_f32_16x16x4_f32(false, a, false, b,
                                                     (short)0, acc0, false, false);
      else
        acc1 = __builtin_amdgcn_wmma_f32_16x16x4_f32(false, a, false, b,
                                                     (short)0, acc1, false, false);
    }
  }
  v8f acc = acc0 + acc1;

  // D layout: lane l in [0,16): VGPR v holds D[v][l] (rows 8-15 are zero, skipped)
  if (lane < 16) {
#pragma unroll
    for (int v = 0; v < 8; ++v) s_D[w][v*16 + lane] = acc[v];
  }
  __syncthreads();

  // ---- Per-item scalar tail on lane 0 of each wave ----
  if (lane == 0) {
    const float* D = &s_D[w][0];
    Cx ph1[2][8], ph2[2][8];
#pragma unroll
    for (int n = 0; n < 2; ++n)
#pragma unroll
      for (int t = 0; t < 8; ++t) {
        ph1[n][t].r = D[ n   *16 + t] - D[(n+2)*16 + t + 8];
        ph1[n][t].i = D[ n   *16 + t + 8] + D[(n+2)*16 + t];
        ph2[n][t].r = D[(n+4)*16 + t] - D[(n+6)*16 + t + 8];
        ph2[n][t].i = D[(n+4)*16 + t + 8] + D[(n+6)*16 + t];
      }

    float f1r[16], f1i[16], f2r[16], f2i[16];
    float s1 = 0.f, s2 = 0.f;
#pragma unroll
    for (int j = 0; j < 16; ++j) {
      f1r[j] = th[200 + j]; f1i[j] = th[216 + j];
      f2r[j] = th[232 + j]; f2i[j] = th[248 + j];
      s1 += f1r[j]*f1r[j] + f1i[j]*f1i[j];
      s2 += f2r[j]*f2r[j] + f2i[j]*f2i[j];
    }
    const float n1 = sqrtf(2.f / s1);
    const float n2 = sqrtf(2.f / s2);
#pragma unroll
    for (int j = 0; j < 16; ++j) {
      f1r[j] *= n1; f1i[j] *= n1;
      f2r[j] *= n2; f2i[j] *= n2;
    }

    Cx A1[2][2], C1[2][2], A2[2][2], C2[2][2];
    phi_times_F(ph1, f1r, f1i, A1);
    phi_times_F(ph1, f2r, f2i, C1);
    phi_times_F(ph2, f2r, f2i, A2);
    phi_times_F(ph2, f1r, f1i, C2);

    const float R1 = rate2(A1, C1);
    const float R2 = rate2(A2, C2);
    s_red[w] = fmaxf(-R1, -R2);
  }
  __syncthreads();

  if (tid == 0) {
    float bs = 0.f;
#pragma unroll
    for (int p = 0; p < ITEMS_PER_BLOCK; ++p) bs += s_red[p];
    block_sums[blockIdx.x] = bs;
  }
}

__global__ void __launch_bounds__(256)
reduce_kernel(const float* __restrict__ block_sums, float* __restrict__ out,
              int n, float scale) {
  __shared__ float sm[256];
  float s = 0.f;
  for (int i = threadIdx.x; i < n; i += 256) s += block_sums[i];
  sm[threadIdx.x] = s;
  __syncthreads();
  for (int off = 128; off > 0; off >>= 1) {
    if ((int)threadIdx.x < off) sm[threadIdx.x] += sm[threadIdx.x + off];
    __syncthreads();
  }
  if (threadIdx.x == 0) out[0] = sm[0] * scale;
}

extern "C" void kernel_launch(void* const* d_in, const int* in_sizes, int n_in,
                              void* d_out, int out_size, void* d_ws, size_t ws_size,
                              hipStream_t stream) {
  const float* theta1 = (const float*)d_in[0];
  const float* g1r    = (const float*)d_in[1];
  const float* g1i    = (const float*)d_in[2];
  const float* g2r    = (const float*)d_in[3];
  const float* g2i    = (const float*)d_in[4];
  const float* ur     = (const float*)d_in[5];
  const float* ui     = (const float*)d_in[6];

  const int n_items = in_sizes[0] / 264;            // 65536
  const int blocks  = n_items / ITEMS_PER_BLOCK;    // 16384
  float* block_sums = (float*)d_ws;

  rate_loss_kernel<<<blocks, NTHREADS, 0, stream>>>(theta1, g1r, g1i, g2r, g2i,
                                                    ur, ui, block_sums);
  reduce_kernel<<<1, 256, 0, stream>>>(block_sums, (float*)d_out, blocks,
                                       1.f / (float)n_items);
}